// LocalAutoCorr2DCUDA_20194936226265
// MI455X (gfx1250) — compile-verified
//
#include <hip/hip_runtime.h>
#include <stdint.h>

typedef float v2f  __attribute__((ext_vector_type(2)));
typedef float v8f  __attribute__((ext_vector_type(8)));
typedef unsigned int u32x4 __attribute__((ext_vector_type(4)));
typedef int i32x4 __attribute__((ext_vector_type(4)));
typedef int i32x8 __attribute__((ext_vector_type(8)));

#define IMG_H   96
#define IMG_W   96
#define NHW     23                 // (96-8)/4+1
#define NWIN    (NHW*NHW)          // 529 windows per (b,c) slice
#define NPAIR   ((NWIN+1)/2)       // 265 patch pairs
#define WAVES   8                  // 256 threads / wave32
#define PSPLIT  4                  // pair-splits per slice (grid.y)
#define IMG_FLOATS (IMG_H*IMG_W)   // 9216
#define DSC_FLOATS 256             // one 16x16 f32 D tile per wave
#define LDS_BYTES ((IMG_FLOATS + WAVES*DSC_FLOATS) * 4)  // 45,056 B

__global__ __launch_bounds__(256)
void autocorr_wmma_f32(const float* __restrict__ x, float* __restrict__ out)
{
    extern __shared__ float lds[];
    float* img    = lds;                       // 9216 floats: whole 96x96 slice
    float* dscAll = lds + IMG_FLOATS;          // 256 floats per wave

    const int bc   = blockIdx.x;               // 0..511  (b*64+c)
    const int tid  = threadIdx.x;
    const int lane = tid & 31;
    const int wave = tid >> 5;

    const float* src = x   + (size_t)bc * IMG_FLOATS;
    float*       dst = out + (size_t)bc * (size_t)NWIN * 64;

    // ---- TDM: one tensor op stages the whole 36KB slice into LDS ----------
    if (wave == 0) {
        const unsigned int ldsOff = __builtin_amdgcn_groupstaticsize(); // dyn-LDS base
        const unsigned long long ga = (unsigned long long)(uintptr_t)src;
        u32x4 g0;
        g0[0] = 1u;                                            // count=1 (valid user D#)
        g0[1] = ldsOff;                                        // lds_addr
        g0[2] = (unsigned int)(ga & 0xFFFFFFFFull);            // global_addr[31:0]
        g0[3] = (unsigned int)((ga >> 32) & 0x01FFFFFFull)     // global_addr[56:32]
              | 0x80000000u;                                   // type=2 ("image")
        i32x8 g1;
        g1[0] = 0x00020000;    // workgroup_mask=0, data_size=2 (4B)
        g1[1] = 0x24000000;    // tensor_dim0 = 9216 (bits 79:48, low half here)
        g1[2] = 0x00010000;    // tensor_dim0 hi=0, tensor_dim1 = 1
        g1[3] = 0x24000000;    // tensor_dim1 hi=0, tile_dim0 = 9216
        g1[4] = 0x00000001;    // tile_dim1 = 1, tile_dim2 = 0
        g1[5] = 9216;          // tensor_dim0_stride (elements)
        g1[6] = 0x24000000;    // tensor_dim1_stride = 9216
        g1[7] = 0;
        i32x4 g2 = {0, 0, 0, 0};
        i32x4 g3 = {0, 0, 0, 0};
        i32x8 g4 = {0, 0, 0, 0, 0, 0, 0, 0};   // extra group (clang-23 6-arg form)
        __builtin_amdgcn_tensor_load_to_lds(g0, g1, g2, g3, g4, 0);
        __builtin_amdgcn_s_wait_tensorcnt(0);
    }
    __syncthreads();

    float* dsc = dscAll + wave * DSC_FLOATS;

    // Per-lane WMMA operand roles (A is 16x4: M=lane%16, K base 0 or 2;
    // B is 4x16: N=lane%16, same K base — per ISA 32-bit A/B striping).
    const int mn  = lane & 15;          // A row m == B col n
    const int qAB = mn >> 3;            // which patch of the pair
    const int rAB = mn & 7;             // row inside the 8x8 patch
    const int ks  = (lane >> 4) << 1;   // lanes 0-15 -> K{0,1}, 16-31 -> K{2,3}

    // Gather role: combo c = lane&15 -> (q, dy); lanes 16..31 are redundant.
    const int cq  = (lane & 15) >> 3;
    const int cdy = lane & 7;
    const int csy = cdy - 4;
    const int gBase = 136 * cq + csy;   // dsc index at a=0: (8cq)*16 + (0+csy+8cq)

    for (int pi = blockIdx.y * WAVES + wave; pi < NPAIR; pi += WAVES * PSPLIT) {
        int w0 = 2 * pi;
        int w1 = 2 * pi + 1;
        bool dup = false;
        if (w1 >= NWIN) { w1 = NWIN - 1; dup = true; }   // odd count: duplicate last

        // This lane's patch row base inside the LDS image (shared by A and B roles)
        const int wq0     = qAB ? w1 : w0;
        const int wy      = wq0 / NHW;
        const int wx      = wq0 % NHW;
        const int rowBase = (wy * 4 + rAB) * IMG_W + wx * 4;

        // A operand (independent of shift): two K=4 slices of columns 0..7
        v2f a_s0, a_s1;
        a_s0.x = img[rowBase + ks + 0];
        a_s0.y = img[rowBase + ks + 1];
        a_s1.x = img[rowBase + ks + 4];
        a_s1.y = img[rowBase + ks + 5];

        float vout[8];                    // out[cdy, dx] for dx = 0..7

        #pragma unroll
        for (int dxi = 0; dxi < 8; ++dxi) {
            const int sx = dxi - 4;

            // B[k=j, n=b] = P[b, j+sx], zero outside [0,8).
            // Load from the clamped (always-valid) in-row address, then select.
            v2f b_s0, b_s1;
            int c; float t;
            c = ks + 0 + sx; t = img[rowBase + (c & 7)]; b_s0.x = ((unsigned)c < 8u) ? t : 0.f;
            c = ks + 1 + sx; t = img[rowBase + (c & 7)]; b_s0.y = ((unsigned)c < 8u) ? t : 0.f;
            c = ks + 4 + sx; t = img[rowBase + (c & 7)]; b_s1.x = ((unsigned)c < 8u) ? t : 0.f;
            c = ks + 5 + sx; t = img[rowBase + (c & 7)]; b_s1.y = ((unsigned)c < 8u) ? t : 0.f;

            // D = A x B over K=8 (two chained K=4 WMMAs), f32 in / f32 acc
            v8f acc = {};
            acc = __builtin_amdgcn_wmma_f32_16x16x4_f32(false, a_s0, false, b_s0,
                                                        (short)0, acc, false, false);
            acc = __builtin_amdgcn_wmma_f32_16x16x4_f32(false, a_s1, false, b_s1,
                                                        (short)0, acc, false, false);

            // Spill the 16x16 D tile to wave-private LDS scratch
            const int Mbase = (lane < 16) ? 0 : 8;
            const int Ncol  = lane & 15;
            #pragma unroll
            for (int r = 0; r < 8; ++r)
                dsc[(Mbase + r) * 16 + Ncol] = acc[r];

            asm volatile("s_wait_dscnt 0" ::: "memory");

            // Diagonal sum: out[sy,sx] = sum_a M_sx[a, a+sy], fully unrolled,
            // predicated (clamped LDS index is always inside the 256-float tile).
            float v = 0.f;
            #pragma unroll
            for (int a = 0; a < 8; ++a) {
                const float d = dsc[(gBase + 17 * a) & 255];
                v += ((unsigned)(a + csy) < 8u) ? d : 0.f;
            }
            vout[dxi] = v;

            asm volatile("" ::: "memory");   // keep next tile's stores after these reads
        }

        // Coalesced write: lanes 0..15 each own one (q,dy) row of 8 dx values
        if (lane < 16 && !(cq && dup)) {
            const int wq = cq ? w1 : w0;
            float* p = dst + (size_t)wq * 64 + cdy * 8;
            reinterpret_cast<float4*>(p)[0] = make_float4(vout[0], vout[1], vout[2], vout[3]);
            reinterpret_cast<float4*>(p)[1] = make_float4(vout[4], vout[5], vout[6], vout[7]);
        }
    }
}

extern "C" void kernel_launch(void* const* d_in, const int* in_sizes, int n_in,
                              void* d_out, int out_size, void* d_ws, size_t ws_size,
                              hipStream_t stream) {
    (void)in_sizes; (void)n_in; (void)d_ws; (void)ws_size; (void)out_size;
    const float* x   = (const float*)d_in[0];
    float*       out = (float*)d_out;
    dim3 grid(8 * 64, PSPLIT);     // one x-slice per block.x, pairs split over block.y
    dim3 block(32 * WAVES);
    autocorr_wmma_f32<<<grid, block, LDS_BYTES, stream>>>(x, out);
}